// Involution_M_45913200394457
// MI455X (gfx1250) — compile-verified
//
#include <hip/hip_runtime.h>
#include <hip/hip_bf16.h>

// ---------------------------------------------------------------------------
// Involution (B=4, C=256, H=W=112, K=7, stride=2, groups=16, gc=16, Cr=64)
// ---------------------------------------------------------------------------
#define BATCH 4
#define CH    256
#define HIN   112
#define WIN   112
#define HO    56
#define WO    56
#define HW    (HO * WO)          // 3136
#define CR    64                 // C / reduction_ratio
#define KK    7
#define K2    49
#define NG    16                 // groups
#define GC    16                 // group channels
#define KO    (K2 * NG)          // 784
#define PAD   3
#define BN_EPS 1e-5f

typedef __attribute__((ext_vector_type(2))) float v2f;
typedef __attribute__((ext_vector_type(8))) float v8f;

// ---------------------------------------------------------------------------
// k1: invden[b][p] = 1 / (avg2x2(m) + 1e-7)
// ---------------------------------------------------------------------------
__global__ void k_invden(const float* __restrict__ m, float* __restrict__ invden) {
    int idx = blockIdx.x * blockDim.x + threadIdx.x;      // [0, BATCH*HW)
    if (idx >= BATCH * HW) return;
    int wo = idx % WO;
    int ho = (idx / WO) % HO;
    int b  = idx / HW;
    const float* mb = m + (size_t)b * HIN * WIN;
    int r0 = (2 * ho) * WIN + 2 * wo;
    float s = mb[r0] + mb[r0 + 1] + mb[r0 + WIN] + mb[r0 + WIN + 1];
    invden[idx] = 1.0f / (0.25f * s + 1e-7f);
}

// ---------------------------------------------------------------------------
// k2: xp[b][c][p] = avg2x2(x) * invden[b][p]
// ---------------------------------------------------------------------------
__global__ void k_maskpool(const float* __restrict__ x, const float* __restrict__ invden,
                           float* __restrict__ xp) {
    int idx = blockIdx.x * blockDim.x + threadIdx.x;      // [0, BATCH*CH*HW)
    if (idx >= BATCH * CH * HW) return;
    int wo = idx % WO;
    int ho = (idx / WO) % HO;
    int c  = (idx / HW) % CH;
    int b  = idx / (HW * CH);
    const float* xb = x + ((size_t)(b * CH + c)) * HIN * WIN;
    int r0 = (2 * ho) * WIN + 2 * wo;
    float s = xb[r0] + xb[r0 + 1] + xb[r0 + WIN] + xb[r0 + WIN + 1];
    xp[idx] = 0.25f * s * invden[b * HW + ho * WO + wo];
}

// ---------------------------------------------------------------------------
// k3: conv1  h = relu(bn(w1 @ xp))   M=64, K=256, N=12544
//     Wave tile: 32M x 32N -> 4 accumulators, 2x2 register blocking.
//     Each A/B fragment feeds 2 WMMAs. 784 wave-tiles, 196 blocks x 128.
// ---------------------------------------------------------------------------
__global__ void k_conv1_wmma(const float* __restrict__ w1, const float* __restrict__ xp,
                             const float* __restrict__ gmm, const float* __restrict__ beta,
                             const float* __restrict__ mean, const float* __restrict__ var,
                             float* __restrict__ h) {
    const int wave = threadIdx.x >> 5;
    const int lane = threadIdx.x & 31;
    const int half = lane >> 4;             // A/B: 0 -> K=0,1 | 1 -> K=2,3
    const int lm   = lane & 15;

    const int tile  = blockIdx.x * 4 + wave;   // [0, 784)
    const int m0    = (tile & 1) * 32;         // 2 M-tiles of 32
    const int nt    = tile >> 1;               // [0, 392)
    const int batch = nt / (HW / 32);          // 98 N-tiles of 32 per batch
    const int pix0  = (nt % (HW / 32)) * 32;

    const float* A0 = w1 + (size_t)(m0 + lm) * CH + half * 2;        // rows m0..m0+15
    const float* A1 = A0 + (size_t)16 * CH;                          // rows m0+16..+31
    const float* Bb = xp + (size_t)batch * CH * HW + pix0 + lm;

    v8f c00 = {}, c01 = {}, c10 = {}, c11 = {};
#pragma unroll 4
    for (int k0 = 0; k0 < CH; k0 += 4) {
        v2f a0, a1, b0, b1;
        a0.x = A0[k0];  a0.y = A0[k0 + 1];
        a1.x = A1[k0];  a1.y = A1[k0 + 1];
        const float* bp = Bb + (size_t)(k0 + half * 2) * HW;
        b0.x = bp[0];   b0.y = bp[HW];
        b1.x = bp[16];  b1.y = bp[HW + 16];
        c00 = __builtin_amdgcn_wmma_f32_16x16x4_f32(false, a0, false, b0, (short)0, c00, false, false);
        c01 = __builtin_amdgcn_wmma_f32_16x16x4_f32(false, a0, false, b1, (short)0, c01, false, false);
        c10 = __builtin_amdgcn_wmma_f32_16x16x4_f32(false, a1, false, b0, (short)0, c10, false, false);
        c11 = __builtin_amdgcn_wmma_f32_16x16x4_f32(false, a1, false, b1, (short)0, c11, false, false);
    }

    // epilogue: BN (eval) + ReLU; row = m0 + v + 8*half (+16), col = pix0 + lm (+16)
    float* out = h + (size_t)batch * CR * HW + pix0 + lm;
#pragma unroll
    for (int v = 0; v < 8; ++v) {
        int r0 = m0 + v + half * 8;
        int r1 = r0 + 16;
        float sc0 = gmm[r0] * __frsqrt_rn(var[r0] + BN_EPS);
        float bi0 = beta[r0] - mean[r0] * sc0;
        float sc1 = gmm[r1] * __frsqrt_rn(var[r1] + BN_EPS);
        float bi1 = beta[r1] - mean[r1] * sc1;
        float v00 = c00[v] * sc0 + bi0;
        float v01 = c01[v] * sc0 + bi0;
        float v10 = c10[v] * sc1 + bi1;
        float v11 = c11[v] * sc1 + bi1;
        out[(size_t)r0 * HW]      = v00 > 0.0f ? v00 : 0.0f;
        out[(size_t)r0 * HW + 16] = v01 > 0.0f ? v01 : 0.0f;
        out[(size_t)r1 * HW]      = v10 > 0.0f ? v10 : 0.0f;
        out[(size_t)r1 * HW + 16] = v11 > 0.0f ? v11 : 0.0f;
    }
}

// ---------------------------------------------------------------------------
// k4: conv2  wt = w2 @ h   M=784, K=64, N=12544
//     Wave tile: 16M x 32N -> 2 accumulators (A fragment reused twice).
//     49 * 392 = 19208 wave-tiles, 4802 blocks x 128.
// ---------------------------------------------------------------------------
__global__ void k_conv2_wmma(const float* __restrict__ w2, const float* __restrict__ h,
                             float* __restrict__ wt) {
    const int wave = threadIdx.x >> 5;
    const int lane = threadIdx.x & 31;
    const int half = lane >> 4;
    const int lm   = lane & 15;

    const int tile  = blockIdx.x * 4 + wave;   // [0, 19208)
    const int m0    = (tile % 49) * 16;
    const int nt    = tile / 49;               // [0, 392)
    const int batch = nt / (HW / 32);
    const int pix0  = (nt % (HW / 32)) * 32;

    const float* A  = w2 + (size_t)(m0 + lm) * CR + half * 2;
    const float* Bb = h + (size_t)batch * CR * HW + pix0 + lm;

    v8f c0 = {}, c1 = {};
#pragma unroll 4
    for (int k0 = 0; k0 < CR; k0 += 4) {
        v2f a, b0, b1;
        a.x = A[k0];    a.y = A[k0 + 1];
        const float* bp = Bb + (size_t)(k0 + half * 2) * HW;
        b0.x = bp[0];   b0.y = bp[HW];
        b1.x = bp[16];  b1.y = bp[HW + 16];
        c0 = __builtin_amdgcn_wmma_f32_16x16x4_f32(false, a, false, b0, (short)0, c0, false, false);
        c1 = __builtin_amdgcn_wmma_f32_16x16x4_f32(false, a, false, b1, (short)0, c1, false, false);
    }

    float* out = wt + (size_t)batch * KO * HW + pix0 + lm;
#pragma unroll
    for (int v = 0; v < 8; ++v) {
        int row = m0 + v + half * 8;
        out[(size_t)row * HW]      = c0[v];
        out[(size_t)row * HW + 16] = c1[v];
    }
}

// ---------------------------------------------------------------------------
// k5: involution gather.
//  Block = (batch, group, 16-pixel tile); 256 thr = 16 chan x 16 pix.
//  Stage the 49x16 per-pixel weight tile in LDS (shared by all 16 channels).
// ---------------------------------------------------------------------------
__global__ void k_involution(const float* __restrict__ x, const float* __restrict__ wt,
                             float* __restrict__ out) {
    __shared__ float wsh[K2 * 16];             // 3136 B

    const int pt   = blockIdx.x % (HW / 16);   // 196 pixel tiles
    const int g    = (blockIdx.x / (HW / 16)) % NG;
    const int b    = blockIdx.x / ((HW / 16) * NG);
    const int pix0 = pt * 16;

    // cooperative weight load: wt[b, g*49 + kk, pix0 + p] -> wsh[kk*16 + p]
    const float* wsrc = wt + ((size_t)b * KO + (size_t)g * K2) * HW + pix0;
    for (int i = threadIdx.x; i < K2 * 16; i += 256) {
        int kk = i >> 4;
        int p  = i & 15;
        wsh[i] = wsrc[(size_t)kk * HW + p];
    }
    __syncthreads();

    const int tp  = threadIdx.x & 15;          // pixel within tile
    const int c   = threadIdx.x >> 4;          // channel within group
    const int pix = pix0 + tp;
    const int wo  = pix % WO;
    const int ho  = pix / WO;
    const int ch  = g * GC + c;

    const float* xb = x + ((size_t)(b * CH + ch)) * HIN * WIN;

    float acc = 0.0f;
#pragma unroll
    for (int ki = 0; ki < KK; ++ki) {
        int hi = 2 * ho - PAD + ki;
        if (hi < 0 || hi >= HIN) continue;
#pragma unroll
        for (int kj = 0; kj < KK; ++kj) {
            int wi = 2 * wo - PAD + kj;
            if (wi < 0 || wi >= WIN) continue;
            acc += wsh[(ki * KK + kj) * 16 + tp] * xb[hi * WIN + wi];
        }
    }
    out[((size_t)(b * CH + ch)) * HW + pix] = acc;
}

// ---------------------------------------------------------------------------
// launch
// ---------------------------------------------------------------------------
extern "C" void kernel_launch(void* const* d_in, const int* in_sizes, int n_in,
                              void* d_out, int out_size, void* d_ws, size_t ws_size,
                              hipStream_t stream) {
    const float* x     = (const float*)d_in[0];
    const float* m     = (const float*)d_in[1];
    const float* w1    = (const float*)d_in[2];
    const float* gmm   = (const float*)d_in[3];
    const float* beta  = (const float*)d_in[4];
    const float* mean  = (const float*)d_in[5];
    const float* var   = (const float*)d_in[6];
    const float* w2    = (const float*)d_in[7];
    float* out = (float*)d_out;

    // workspace layout (bytes)
    char* ws = (char*)d_ws;
    float* xp     = (float*)(ws);                                   // 4*256*3136*4 = 12,845,056
    float* invden = (float*)(ws + 12845056);                        // 12544*4      =     50,176
    float* h      = (float*)(ws + 12845056 + 50176);                // 4*64*3136*4  =  3,211,264
    float* wt     = (float*)(ws + 12845056 + 50176 + 3211264);      // 4*784*3136*4 = 39,337,984

    // k1: pooled-mask reciprocal
    k_invden<<<(BATCH * HW + 255) / 256, 256, 0, stream>>>(m, invden);

    // k2: masked average pool
    k_maskpool<<<(BATCH * CH * HW + 255) / 256, 256, 0, stream>>>(x, invden, xp);

    // k3: conv1 + BN + ReLU  (2x392 = 784 wave-tiles of 32Mx32N)
    k_conv1_wmma<<<784 / 4, 128, 0, stream>>>(w1, xp, gmm, beta, mean, var, h);

    // k4: conv2 -> per-pixel involution kernels (49*392 = 19208 wave-tiles of 16Mx32N)
    k_conv2_wmma<<<19208 / 4, 128, 0, stream>>>(w2, h, wt);

    // k5: involution gather + reduce (weights staged in LDS per group)
    k_involution<<<BATCH * NG * (HW / 16), 256, 0, stream>>>(x, wt, out);
}